// TypeAttention_10685878632690
// MI455X (gfx1250) — compile-verified
//
#include <hip/hip_runtime.h>

#define N_P 100000
#define N_A 50000
#define DIM 64
#define E_W 800000
#define E_C 1200000

typedef __attribute__((ext_vector_type(2))) float v2f;
typedef __attribute__((ext_vector_type(8))) float v8f;

// ---------------- workspace layout (floats) ----------------
// [0)            rst_w  -> att_w   N_P*64
// [6.4M)         rst_c  -> att_c   N_P*64
// [12.8M)        outdeg_author     N_A   (-> rsqrt(max(.,1)) in place)
// [12.85M)       outdeg_paper      N_P   (-> rsqrt in place)
// [12.95M)       cnt_w (in-deg writes, raw)  N_P
// [13.05M)       cnt_c (in-deg cites, raw)   N_P
static const size_t OFF_RSTW = 0;
static const size_t OFF_RSTC = (size_t)N_P * DIM;
static const size_t OFF_ODA  = OFF_RSTC + (size_t)N_P * DIM;
static const size_t OFF_ODP  = OFF_ODA + N_A;
static const size_t OFF_CW   = OFF_ODP + N_P;
static const size_t OFF_CC   = OFF_CW + N_P;
static const size_t WS_FLOATS = OFF_CC + N_P;   // 13,150,000 floats = 52.6 MB

// ---------------------------------------------------------------------------
__global__ void zero_f32(float* __restrict__ p, int n) {
  int i = blockIdx.x * blockDim.x + threadIdx.x;
  if (i < n) p[i] = 0.0f;
}

// one thread per edge: histogram src and dst degrees
__global__ void edge_degrees(const int* __restrict__ src, const int* __restrict__ dst,
                             int ne, float* __restrict__ deg_src, float* __restrict__ deg_dst) {
  int e = blockIdx.x * blockDim.x + threadIdx.x;
  if (e >= ne) return;
  atomicAdd(&deg_src[src[e]], 1.0f);
  atomicAdd(&deg_dst[dst[e]], 1.0f);
}

// in place: d = rsqrt(max(d,1))
__global__ void rsqrt_clip(float* __restrict__ p, int n) {
  int i = blockIdx.x * blockDim.x + threadIdx.x;
  if (i < n) p[i] = rsqrtf(fmaxf(p[i], 1.0f));
}

// 16 threads per edge, float4 gather of source row * invsqrt(outdeg), scatter-add to dst row
__global__ void scatter_norm_add(const float* __restrict__ hsrc,
                                 const int* __restrict__ src, const int* __restrict__ dst,
                                 const float* __restrict__ inv_sqrt_out,
                                 float* __restrict__ rst, int ne) {
  int t = blockIdx.x * blockDim.x + threadIdx.x;
  int e = t >> 4;
  if (e >= ne) return;
  int q = (t & 15) * 4;
  int s  = src[e];
  int d0 = dst[e];
  float sc = inv_sqrt_out[s];
  float4 v = *(const float4*)(hsrc + (size_t)s * DIM + q);
  float* out = rst + (size_t)d0 * DIM + q;
  atomicAdd(out + 0, v.x * sc);
  atomicAdd(out + 1, v.y * sc);
  atomicAdd(out + 2, v.z * sc);
  atomicAdd(out + 3, v.w * sc);
}

// ---------------------------------------------------------------------------
// Fused: in-degree normalize -> 3x GEMM (WMMA f32 16x16x4) -> bias -> elu ->
// closed-form heterogeneous edge softmax per node. One block = 16 paper rows,
// 4 waves = 4 column tiles of 16. Writes att_w/att_c in place over rst_w/rst_c.
#define LP 68   // LDS row pitch (floats); stride-4 banks per lane -> conflict-free frags

__global__ __launch_bounds__(128) void fused_transform(
    const float* __restrict__ h_paper,
    float* __restrict__ rst_w, float* __restrict__ rst_c,
    const float* __restrict__ Wl,  const float* __restrict__ bl,
    const float* __restrict__ Wra, const float* __restrict__ bra,
    const float* __restrict__ Wrp, const float* __restrict__ brp,
    const float* __restrict__ cnt_w, const float* __restrict__ cnt_c) {
  __shared__ float sAl[16 * LP], sAw[16 * LP], sAc[16 * LP];
  __shared__ float sCW[16], sCC[16], sSW[16], sSC[16];

  const int tid  = threadIdx.x;
  const int row0 = blockIdx.x * 16;

  if (tid < 16) {
    float cw = cnt_w[row0 + tid];
    float cc = cnt_c[row0 + tid];
    sCW[tid] = cw;  sCC[tid] = cc;
    sSW[tid] = rsqrtf(fmaxf(cw, 1.0f));   // in-degree normalization (writes rel)
    sSC[tid] = rsqrtf(fmaxf(cc, 1.0f));   // in-degree normalization (cites rel)
  }
  __syncthreads();

  for (int i = tid; i < 16 * 64; i += 128) {
    int r = i >> 6, c = i & 63;
    size_t g = (size_t)(row0 + r) * DIM + c;
    sAl[r * LP + c] = h_paper[g];
    sAw[r * LP + c] = rst_w[g] * sSW[r];
    sAc[r * LP + c] = rst_c[g] * sSC[r];
  }
  __syncthreads();

  const int lane = tid & 31;
  const int wave = tid >> 5;
  const int hlf  = lane >> 4;        // A: K half / C-D: M half
  const int lm   = lane & 15;        // A: row M / B: column within tile
  const int ncol = wave * 16 + lm;   // output column 0..63

  v8f accL = {}, accW = {}, accC = {};

  #pragma unroll
  for (int k = 0; k < 16; ++k) {
    const int kk = 4 * k + 2 * hlf;
    // A fragments from LDS: lane<16 holds (M=lm, K=kk,kk+1); lane>=16 holds K=kk,kk+1 with kk=4k+2
    v2f al, aw, ac;
    al.x = sAl[lm * LP + kk]; al.y = sAl[lm * LP + kk + 1];
    aw.x = sAw[lm * LP + kk]; aw.y = sAw[lm * LP + kk + 1];
    ac.x = sAc[lm * LP + kk]; ac.y = sAc[lm * LP + kk + 1];
    // B[kq, n] = W[n, kq]  (h @ W^T); W row-major -> contiguous pair per lane
    v2f b0, b1, b2;
    const float* wl = Wl  + (size_t)ncol * DIM + kk;  b0.x = wl[0]; b0.y = wl[1];
    const float* wa = Wra + (size_t)ncol * DIM + kk;  b1.x = wa[0]; b1.y = wa[1];
    const float* wp = Wrp + (size_t)ncol * DIM + kk;  b2.x = wp[0]; b2.y = wp[1];

    accL = __builtin_amdgcn_wmma_f32_16x16x4_f32(false, al, false, b0, (short)0, accL, false, false);
    accW = __builtin_amdgcn_wmma_f32_16x16x4_f32(false, aw, false, b1, (short)0, accW, false, false);
    accC = __builtin_amdgcn_wmma_f32_16x16x4_f32(false, ac, false, b2, (short)0, accC, false, false);
  }

  const float biasL = bl[ncol], biasA = bra[ncol], biasP = brp[ncol];

  #pragma unroll
  for (int i = 0; i < 8; ++i) {
    const int mrow = i + 8 * hlf;    // C/D layout: VGPR i -> M=i (lanes 0-15), M=i+8 (lanes 16-31)
    float hl = accL[i] + biasL;
    float pw = hl + accW[i] + biasA;
    float pc = hl + accC[i] + biasP;
    float eaw = pw > 0.0f ? pw : (__expf(pw) - 1.0f);   // elu
    float eac = pc > 0.0f ? pc : (__expf(pc) - 1.0f);

    float cw = sCW[mrow], cc = sCC[mrow];
    bool hw = cw > 0.0f, hc = cc > 0.0f;
    float mval = fmaxf(hw ? eaw : -3.402823466e38f, hc ? eac : -3.402823466e38f);
    float xw = hw ? __expf(eaw - mval) : 0.0f;
    float xc = hc ? __expf(eac - mval) : 0.0f;
    float z  = cw * xw + cc * xc;
    float inv = (hw || hc) ? (1.0f / z) : 0.0f;

    size_t g = (size_t)(row0 + mrow) * DIM + ncol;
    rst_w[g] = xw * inv;   // att_w per node
    rst_c[g] = xc * inv;   // att_c per node
  }
}

// out[e,:] = att[dst[e],:]  (16 threads/edge, float4 -> 256B coalesced store per edge)
__global__ void gather_out(const float* __restrict__ att, const int* __restrict__ dst,
                           float* __restrict__ out, int ne) {
  int t = blockIdx.x * blockDim.x + threadIdx.x;
  int e = t >> 4;
  if (e >= ne) return;
  int q = (t & 15) * 4;
  int d0 = dst[e];
  *(float4*)(out + (size_t)e * DIM + q) = *(const float4*)(att + (size_t)d0 * DIM + q);
}

// ---------------------------------------------------------------------------
extern "C" void kernel_launch(void* const* d_in, const int* in_sizes, int n_in,
                              void* d_out, int out_size, void* d_ws, size_t ws_size,
                              hipStream_t stream) {
  const float* h_paper  = (const float*)d_in[0];
  const float* h_author = (const float*)d_in[1];
  const float* Wl       = (const float*)d_in[2];
  const float* bl       = (const float*)d_in[3];
  const float* Wra      = (const float*)d_in[4];
  const float* bra      = (const float*)d_in[5];
  const float* Wrp      = (const float*)d_in[6];
  const float* brp      = (const float*)d_in[7];
  const int* wsrc = (const int*)d_in[8];
  const int* wdst = (const int*)d_in[9];
  const int* csrc = (const int*)d_in[10];
  const int* cdst = (const int*)d_in[11];

  float* ws   = (float*)d_ws;
  float* rstw = ws + OFF_RSTW;
  float* rstc = ws + OFF_RSTC;
  float* oda  = ws + OFF_ODA;
  float* odp  = ws + OFF_ODP;
  float* cw   = ws + OFF_CW;
  float* cc   = ws + OFF_CC;
  float* outp = (float*)d_out;

  const int B = 256;

  // 1) zero accumulators + degree arrays
  zero_f32<<<(int)((WS_FLOATS + B - 1) / B), B, 0, stream>>>(ws, (int)WS_FLOATS);

  // 2) degree histograms
  edge_degrees<<<(E_W + B - 1) / B, B, 0, stream>>>(wsrc, wdst, E_W, oda, cw);
  edge_degrees<<<(E_C + B - 1) / B, B, 0, stream>>>(csrc, cdst, E_C, odp, cc);

  // 3) out-degree -> rsqrt(max(.,1)) in place (oda and odp are contiguous)
  rsqrt_clip<<<((N_A + N_P) + B - 1) / B, B, 0, stream>>>(oda, N_A + N_P);

  // 4) scatter-aggregate both relations
  {
    long long tw = (long long)E_W * 16;
    long long tc = (long long)E_C * 16;
    scatter_norm_add<<<(int)((tw + B - 1) / B), B, 0, stream>>>(h_author, wsrc, wdst, oda, rstw, E_W);
    scatter_norm_add<<<(int)((tc + B - 1) / B), B, 0, stream>>>(h_paper,  csrc, cdst, odp, rstc, E_C);
  }

  // 5) fused WMMA transform + elu + hetero softmax (per-node att tables, in place)
  fused_transform<<<N_P / 16, 128, 0, stream>>>(h_paper, rstw, rstc,
                                                Wl, bl, Wra, bra, Wrp, brp, cw, cc);

  // 6) edge outputs: gathers from per-node tables
  {
    long long tw = (long long)E_W * 16;
    long long tc = (long long)E_C * 16;
    gather_out<<<(int)((tw + B - 1) / B), B, 0, stream>>>(rstw, wdst, outp, E_W);
    gather_out<<<(int)((tc + B - 1) / B), B, 0, stream>>>(rstc, cdst, outp + (size_t)E_W * DIM, E_C);
  }
}